// wave_Layer_74990128988641
// MI455X (gfx1250) — compile-verified
//
#include <hip/hip_runtime.h>

typedef float v2f __attribute__((ext_vector_type(2)));
typedef float v8f __attribute__((ext_vector_type(8)));

#define S_LEN 2048
#define D_DIM 64
#define KMAX  8

// One wave32 per batch row. Sparse formulation of the wavelet/roll pooling:
// each anchor match at s' adds 1/(k*c_j) at position s'+j+1 (if < L); the
// weighted pooling over gathered rows is done with V_WMMA_F32_16X16X4_F32.
// The seqs row is staged to LDS with CDNA5 async memory->LDS loads.
__global__ __launch_bounds__(32) void poi_wave_pool_kernel(
    const int*   __restrict__ seqs,   // [B, S]
    const float* __restrict__ emb,    // [B, S, D]
    const int*   __restrict__ kptr,   // scalar k
    float*       __restrict__ out)    // [B, D]
{
    __shared__ int   sh_seq[S_LEN];
    __shared__ int   sh_ls[S_LEN + 4];   // compacted target positions (+pad)
    __shared__ float sh_lw[S_LEN + 4];   // compacted weights (+pad)

    const int b    = blockIdx.x;
    const int lane = threadIdx.x;        // 0..31

    // ---- stage seqs row into LDS via async global->LDS b128 (ASYNCcnt) ----
    {
        // Low 32 bits of a flat LDS pointer == byte offset in wave LDS space.
        unsigned int lds0 = (unsigned int)(uintptr_t)(void*)sh_seq;
        unsigned long long gbase =
            (unsigned long long)(uintptr_t)(seqs + (size_t)b * S_LEN);
        #pragma unroll
        for (int it = 0; it < (S_LEN * 4) / (32 * 16); ++it) {
            unsigned int off  = (unsigned int)(it * 512 + lane * 16);
            unsigned int ldsa = lds0 + off;
            asm volatile("global_load_async_to_lds_b128 %0, %1, %2"
                         :
                         : "v"(ldsa), "v"(off), "s"(gbase)
                         : "memory");
        }
        asm volatile("s_wait_asynccnt 0" ::: "memory");
    }
    __syncthreads();

    int k = *kptr;
    if (k < 1) k = 1;
    if (k > KMAX) k = KMAX;

    // ---- L = count of valid (nonzero) entries; wave32 butterfly reduce ----
    int cntnz = 0;
    for (int s = lane; s < S_LEN; s += 32) cntnz += (sh_seq[s] != 0);
    for (int m = 16; m >= 1; m >>= 1) cntnz += __shfl_xor(cntnz, m, 32);
    const int L = cntnz;

    // ---- anchors a_j = seq[max(L-1-j,0)] ----
    int anc[KMAX];
    for (int j = 0; j < k; ++j) {
        int ai = L - 1 - j;
        if (ai < 0) ai = 0;
        anc[j] = sh_seq[ai];
    }

    // ---- c_j = #matches of a_j within [0, L) ----
    int cj[KMAX];
    for (int j = 0; j < k; ++j) cj[j] = 0;
    for (int s = lane; s < L; s += 32) {
        int v = sh_seq[s];
        for (int j = 0; j < k; ++j) cj[j] += (v == anc[j]);
    }
    for (int j = 0; j < k; ++j)
        for (int m = 16; m >= 1; m >>= 1) cj[j] += __shfl_xor(cj[j], m, 32);

    float inv[KMAX];
    for (int j = 0; j < k; ++j)
        inv[j] = (cj[j] > 0) ? 1.0f / ((float)k * (float)cj[j]) : 0.0f;

    // ---- deterministic ballot-compaction of nonzero weights w[s], s < L ----
    // w[s] = sum_j [s>=j+1][seq[s-j-1]==a_j] * inv_j
    int nnz = 0;
    for (int sb = 0; sb < L; sb += 32) {
        int s = sb + lane;
        float w = 0.0f;
        if (s < L) {
            for (int j = 0; j < k; ++j) {
                int sp = s - j - 1;
                if (sp >= 0 && sh_seq[sp] == anc[j]) w += inv[j];
            }
        }
        unsigned long long mask = __ballot(w != 0.0f);
        int pre = __popcll(mask & ((1ull << lane) - 1ull));
        if (w != 0.0f) {
            sh_ls[nnz + pre] = s;
            sh_lw[nnz + pre] = w;
        }
        nnz += __popcll(mask);
    }
    // pad list to a multiple of 4 with (s=0, w=0) sentinels -> branch-free loop
    const int nnz4 = (nnz + 3) & ~3;
    if (nnz + lane < nnz4) {             // at most lanes 0..2
        sh_ls[nnz + lane] = 0;
        sh_lw[nnz + lane] = 0.0f;
    }
    __syncthreads();

    // ---- WMMA accumulation: out[b,0:64] = sum_i w_i * emb[b, s_i, 0:64] ----
    // A (16x4 f32): M=0 row = weights, rows 1..15 zero.
    //   lane0: v0=A[0,K0], v1=A[0,K1]; lane16: v0=A[0,K2], v1=A[0,K3].
    // B (4x16 f32), n-tile t covers dims t*16..t*16+15:
    //   v0: lanes0-15 row K0, lanes16-31 row K2; v1: rows K1 / K3.
    // D (16x16 f32): row M=0 = element 0 of lanes 0..15.
    v8f c0 = {}, c1 = {}, c2 = {}, c3 = {};
    const int half = lane >> 4;          // 0 | 1
    const int col  = lane & 15;          // N index within tile
    const float* ebase = emb + (size_t)b * S_LEN * D_DIM;

    for (int i0 = 0; i0 < nnz4; i0 += 4) {
        const int e0 = i0 + 2 * half;    // entry for rows K0/K2 of this half
        const int s0 = sh_ls[e0];        // row for B vgpr0
        const int s1 = sh_ls[e0 + 1];    // row for B vgpr1 (K1/K3)
        const float w0 = sh_lw[e0];
        const float w1 = sh_lw[e0 + 1];
        const float* r0 = ebase + (size_t)s0 * D_DIM;
        const float* r1 = ebase + (size_t)s1 * D_DIM;

        v2f a;
        a.x = (col == 0) ? w0 : 0.0f;
        a.y = (col == 0) ? w1 : 0.0f;

        v2f b0, b1, b2, b3;
        b0.x = r0[ 0 + col];  b0.y = r1[ 0 + col];
        b1.x = r0[16 + col];  b1.y = r1[16 + col];
        b2.x = r0[32 + col];  b2.y = r1[32 + col];
        b3.x = r0[48 + col];  b3.y = r1[48 + col];

        c0 = __builtin_amdgcn_wmma_f32_16x16x4_f32(false, a, false, b0, (short)0, c0, false, false);
        c1 = __builtin_amdgcn_wmma_f32_16x16x4_f32(false, a, false, b1, (short)0, c1, false, false);
        c2 = __builtin_amdgcn_wmma_f32_16x16x4_f32(false, a, false, b2, (short)0, c2, false, false);
        c3 = __builtin_amdgcn_wmma_f32_16x16x4_f32(false, a, false, b3, (short)0, c3, false, false);
    }

    if (lane < 16) {
        float* orow = out + (size_t)b * D_DIM;
        orow[ 0 + lane] = c0[0];
        orow[16 + lane] = c1[0];
        orow[32 + lane] = c2[0];
        orow[48 + lane] = c3[0];
    }
}

extern "C" void kernel_launch(void* const* d_in, const int* in_sizes, int n_in,
                              void* d_out, int out_size, void* d_ws, size_t ws_size,
                              hipStream_t stream) {
    const int*   seqs = (const int*)d_in[0];
    const float* emb  = (const float*)d_in[1];
    const int*   kptr = (const int*)d_in[2];
    float*       out  = (float*)d_out;

    const int B = in_sizes[0] / S_LEN;   // 256
    poi_wave_pool_kernel<<<B, 32, 0, stream>>>(seqs, emb, kptr, out);
}